// PointResnet_49778670961211
// MI455X (gfx1250) — compile-verified
//
#include <hip/hip_runtime.h>

// Problem constants (match reference)
#define B_    8
#define N_    4096
#define KNBR  32
#define CIN   68      // D+3 padded to 68 (multiple of 4 for K-steps)
#define MLP_  64
#define P1_   128
#define P2_   64
#define R2_   0.04f   // RADIUS^2
#define EPS_  1e-5f

typedef float v2f __attribute__((ext_vector_type(2)));
typedef float v8f __attribute__((ext_vector_type(8)));

// D(16x16,f32) = A(16x4,f32) * B(4x16,f32) + C   -- native f32 WMMA on CDNA5
__device__ __forceinline__ v8f wmma4(v2f a, v2f b, v8f c) {
  return __builtin_amdgcn_wmma_f32_16x16x4_f32(false, a, false, b, (short)0, c,
                                               false, false);
}

// ---------------------------------------------------------------------------
// Kernel 0: copy xyz input to first output slot (tuple output: (xyz, out))
// ---------------------------------------------------------------------------
__global__ void copy_xyz_k(const float* __restrict__ xyz, float* __restrict__ out) {
  int i = blockIdx.x * blockDim.x + threadIdx.x;
  if (i < B_ * 3 * N_) out[i] = xyz[i];
}

// ---------------------------------------------------------------------------
// Kernel 1: pack points [B,64,N] + xyz [B,3,N] into pts_t [B][N][68]
//   c<64  : point features
//   64..66: absolute xyz (relative correction folded into GEMM epilogue)
//   67    : zero pad
// ---------------------------------------------------------------------------
__global__ void pack_pts_k(const float* __restrict__ points,
                           const float* __restrict__ xyz,
                           float* __restrict__ pts_t) {
  int idx = blockIdx.x * blockDim.x + threadIdx.x;
  const int total = B_ * N_ * CIN;
  if (idx >= total) return;
  int b = idx / (N_ * CIN);
  int r = idx - b * N_ * CIN;
  int n = r / CIN;
  int c = r - n * CIN;
  float v;
  if (c < 64)      v = points[((size_t)b * 64 + c) * N_ + n];
  else if (c < 67) v = xyz[((size_t)b * 3 + (c - 64)) * N_ + n];
  else             v = 0.f;
  pts_t[idx] = v;
}

// ---------------------------------------------------------------------------
// Kernel 2: ball query + grouped MLP (WMMA) + BN + ReLU + max-over-K
//   1 block = 1 batch x 256 points, 256 threads = 8 waves.
//   Phase 1: thread-per-point index scan (first 32 indices within radius).
//   Phase 2: wave-per-point GEMM [32 x 68] * [68 x 64] via f32 WMMA:
//            2 M-tiles x 4 N-tiles x 17 K-steps.
// ---------------------------------------------------------------------------
__global__ __launch_bounds__(256)
void ball_mlp_k(const float* __restrict__ xyz,
                const float* __restrict__ pts_t,
                const float* __restrict__ W,
                const float* __restrict__ bb, const float* __restrict__ g,
                const float* __restrict__ bt, const float* __restrict__ rm,
                const float* __restrict__ rv,
                float* __restrict__ feat) {
  extern __shared__ float sm[];
  float* sxyz = sm;                      // 3*4096 (x | y | z, whole batch)
  float* Wl   = sm + 3 * N_;             // 64*68 row-major, col 67 zeroed
  float* scl  = Wl + 64 * CIN;           // 64: g*rsqrt(rv+eps)
  float* off  = scl + 64;                // 64: b - rm
  float* btl  = off + 64;                // 64: bt
  int*   giS  = (int*)(btl + 64);        // 256*32 neighbor indices

  const int t        = threadIdx.x;
  const int b        = blockIdx.x >> 4;          // N/256 = 16 tiles per batch
  const int tileBase = (blockIdx.x & 15) * 256;

  // Stage xyz for the whole batch (layout in memory is already [3][N])
  for (int i = t; i < 3 * N_; i += 256) sxyz[i] = xyz[(size_t)b * 3 * N_ + i];
  // Stage W with K padded to 68
  for (int i = t; i < 64 * CIN; i += 256) {
    int o = i / CIN, c = i - o * CIN;
    Wl[i] = (c < 67) ? W[o * 67 + c] : 0.f;
  }
  if (t < 64) {
    float s = g[t] * rsqrtf(rv[t] + EPS_);
    scl[t] = s; off[t] = bb[t] - rm[t]; btl[t] = bt[t];
  }
  __syncthreads();

  const float* sx = sxyz, * sy = sxyz + N_, * sz = sxyz + 2 * N_;

  // ---- Phase 1: ball query (first KNBR indices within radius, in order) ----
  {
    const int n = tileBase + t;
    const float cx = sx[n], cy = sy[n], cz = sz[n];
    int cnt = 0;
    const int base = t * KNBR;
    for (int m = 0; m < N_ && cnt < KNBR; ++m) {   // LDS broadcast reads
      float dx = sx[m] - cx, dy = sy[m] - cy, dz = sz[m] - cz;
      if (dx * dx + dy * dy + dz * dz <= R2_) { giS[base + cnt] = m; ++cnt; }
    }
    const int g0 = giS[base];                       // self is always in-radius
    for (int q = cnt; q < KNBR; ++q) giS[base + q] = g0;
  }
  __syncthreads();

  // ---- Phase 2: per-point WMMA GEMM + epilogue ----
  const int wave = t >> 5, lane = t & 31, l15 = lane & 15;
  const int ks   = (lane >> 4) << 1;               // K sub-offset 0 / 2
  const float* ptsB = pts_t + (size_t)b * N_ * CIN;

  for (int p = wave; p < 256; p += 8) {
    const int n   = tileBase + p;
    const int gi0 = giS[p * KNBR + l15];            // neighbor rows 0..15
    const int gi1 = giS[p * KNBR + 16 + l15];       // neighbor rows 16..31
    const float* r0 = ptsB + (size_t)gi0 * CIN;
    const float* r1 = ptsB + (size_t)gi1 * CIN;
    __builtin_prefetch(r0, 0, 3);                   // global_prefetch_b8
    __builtin_prefetch(r1, 0, 3);

    v8f acc[2][4] = {};
#pragma unroll
    for (int kt = 0; kt < 17; ++kt) {
      const int k0 = kt * 4 + ks;
      v2f a0 = *(const v2f*)(r0 + k0);              // A frag, M-tile 0
      v2f a1 = *(const v2f*)(r1 + k0);              // A frag, M-tile 1
#pragma unroll
      for (int nt = 0; nt < 4; ++nt) {
        v2f bf = *(const v2f*)(&Wl[(nt * 16 + l15) * CIN + k0]);  // B = W^T
        acc[0][nt] = wmma4(a0, bf, acc[0][nt]);
        acc[1][nt] = wmma4(a1, bf, acc[1][nt]);
      }
    }

    // Epilogue: subtract center*W[:,64:67] (rank-1 per-point column bias),
    // BN(eval) + ReLU, max over 32 neighbors (rows).
    const float cx = sx[n], cy = sy[n], cz = sz[n];
#pragma unroll
    for (int nt = 0; nt < 4; ++nt) {
      const int o = nt * 16 + l15;                  // output channel (column)
      const float s  = scl[o];
      const float cc = cx * Wl[o * CIN + 64] + cy * Wl[o * CIN + 65] +
                       cz * Wl[o * CIN + 66];
      const float bias = (off[o] - cc) * s + btl[o];
      float vm = 0.f;                               // ReLU => max >= 0
#pragma unroll
      for (int mt = 0; mt < 2; ++mt)
#pragma unroll
        for (int r = 0; r < 8; ++r)
          vm = fmaxf(vm, acc[mt][nt][r] * s + bias);
      vm = fmaxf(vm, __shfl_xor(vm, 16, 32));       // rows 0-7 vs 8-15 halves
      if (lane < 16) feat[((size_t)b * N_ + n) * 64 + o] = vm;
    }
  }
}

// ---------------------------------------------------------------------------
// Kernel 3: fused resnet  out = relu(BN2(W2 * relu(BN1(W1 * feat))) + points)
//   1 block = 1 batch x 128 rows (points); 8 waves, 16 rows per wave.
//   GEMM1: [16 x 64]*[64 x 128]; Y staged per-wave in LDS (bank-padded 132);
//   GEMM2: [16 x 128]*[128 x 64]; epilogue scatters channel-major output.
// ---------------------------------------------------------------------------
__global__ __launch_bounds__(256)
void resnet_k(const float* __restrict__ feat, const float* __restrict__ points,
              const float* __restrict__ W1, const float* __restrict__ b1,
              const float* __restrict__ g1, const float* __restrict__ bt1,
              const float* __restrict__ rm1, const float* __restrict__ rv1,
              const float* __restrict__ W2, const float* __restrict__ b2,
              const float* __restrict__ g2, const float* __restrict__ bt2,
              const float* __restrict__ rm2, const float* __restrict__ rv2,
              float* __restrict__ out) {
  extern __shared__ float sm[];
  float* W1l = sm;                        // 128*66 (bank-padded)
  float* W2l = W1l + P1_ * 66;            // 64*130 (bank-padded)
  float* s1  = W2l + P2_ * 130;           // 128
  float* c1  = s1 + P1_;                  // 128
  float* s2  = c1 + P1_;                  // 64
  float* c2  = s2 + P2_;                  // 64
  float* Y   = c2 + P2_;                  // 8 waves * 16 rows * stride 132

  const int t       = threadIdx.x;
  const int b       = blockIdx.x >> 5;           // N/128 = 32 tiles per batch
  const int rowBase = (blockIdx.x & 31) * 128;

  for (int i = t; i < P1_ * 66; i += 256) {
    int o = i / 66, c = i - o * 66;
    W1l[i] = (c < 64) ? W1[o * 64 + c] : 0.f;
  }
  for (int i = t; i < P2_ * 130; i += 256) {
    int o = i / 130, c = i - o * 130;
    W2l[i] = (c < 128) ? W2[o * 128 + c] : 0.f;
  }
  if (t < P1_) {
    float s = g1[t] * rsqrtf(rv1[t] + EPS_);
    s1[t] = s; c1[t] = (b1[t] - rm1[t]) * s + bt1[t];
  }
  if (t < P2_) {
    float s = g2[t] * rsqrtf(rv2[t] + EPS_);
    s2[t] = s; c2[t] = (b2[t] - rm2[t]) * s + bt2[t];
  }
  __syncthreads();

  const int wave = t >> 5, lane = t & 31, l15 = lane & 15;
  const int ks   = (lane >> 4) << 1;
  const int mHi  = (lane >> 4) << 3;             // row offset 0 / 8 per half
  const int myRow = rowBase + wave * 16;
  const float* Xrow = feat + ((size_t)b * N_ + myRow + l15) * 64;
  float* Yw = Y + wave * 16 * 132;

  // ---- GEMM1: K=64, N=128 (8 N-tiles) ----
  v8f acc1[8] = {};
#pragma unroll
  for (int kt = 0; kt < 16; ++kt) {
    const int k0 = kt * 4 + ks;
    v2f a = *(const v2f*)(Xrow + k0);
#pragma unroll
    for (int nt = 0; nt < 8; ++nt) {
      v2f bf = *(const v2f*)(&W1l[(nt * 16 + l15) * 66 + k0]);
      acc1[nt] = wmma4(a, bf, acc1[nt]);
    }
  }
  // BN1 + ReLU -> per-wave Y tile in LDS (D-frag -> A-frag layout change)
#pragma unroll
  for (int nt = 0; nt < 8; ++nt) {
    const int o = nt * 16 + l15;
    const float s = s1[o], c = c1[o];
#pragma unroll
    for (int r = 0; r < 8; ++r)
      Yw[(r + mHi) * 132 + o] = fmaxf(acc1[nt][r] * s + c, 0.f);
  }
  __syncthreads();

  // ---- GEMM2: K=128, N=64 (4 N-tiles) ----
  v8f acc2[4] = {};
#pragma unroll
  for (int kt = 0; kt < 32; ++kt) {
    const int k0 = kt * 4 + ks;
    v2f a = *(const v2f*)(&Yw[l15 * 132 + k0]);
#pragma unroll
    for (int nt = 0; nt < 4; ++nt) {
      v2f bf = *(const v2f*)(&W2l[(nt * 16 + l15) * 130 + k0]);
      acc2[nt] = wmma4(a, bf, acc2[nt]);
    }
  }
  // BN2 + residual + ReLU, scatter to channel-major [B,64,N]
#pragma unroll
  for (int nt = 0; nt < 4; ++nt) {
    const int o = nt * 16 + l15;
    const float s = s2[o], c = c2[o];
#pragma unroll
    for (int r = 0; r < 8; ++r) {
      const int n = myRow + r + mHi;
      const size_t idx = ((size_t)b * 64 + o) * N_ + n;
      out[idx] = fmaxf(acc2[nt][r] * s + c + points[idx], 0.f);
    }
  }
}

// ---------------------------------------------------------------------------
extern "C" void kernel_launch(void* const* d_in, const int* in_sizes, int n_in,
                              void* d_out, int out_size, void* d_ws, size_t ws_size,
                              hipStream_t stream) {
  (void)in_sizes; (void)n_in; (void)out_size; (void)ws_size;
  const float* xyz    = (const float*)d_in[0];
  const float* points = (const float*)d_in[1];
  const float* W   = (const float*)d_in[2];
  const float* bb  = (const float*)d_in[3];
  const float* g   = (const float*)d_in[4];
  const float* bt  = (const float*)d_in[5];
  const float* rm  = (const float*)d_in[6];
  const float* rv  = (const float*)d_in[7];
  const float* W1  = (const float*)d_in[8];
  const float* b1  = (const float*)d_in[9];
  const float* g1  = (const float*)d_in[10];
  const float* bt1 = (const float*)d_in[11];
  const float* rm1 = (const float*)d_in[12];
  const float* rv1 = (const float*)d_in[13];
  const float* W2  = (const float*)d_in[14];
  const float* b2  = (const float*)d_in[15];
  const float* g2  = (const float*)d_in[16];
  const float* bt2 = (const float*)d_in[17];
  const float* rm2 = (const float*)d_in[18];
  const float* rv2 = (const float*)d_in[19];

  float* out   = (float*)d_out;
  float* pts_t = (float*)d_ws;                          // B*N*68 floats (8.9 MB)
  float* feat  = pts_t + (size_t)B_ * N_ * CIN;         // B*N*64 floats (8.4 MB)

  copy_xyz_k<<<(B_ * 3 * N_ + 255) / 256, 256, 0, stream>>>(xyz, out);
  pack_pts_k<<<(B_ * N_ * CIN + 255) / 256, 256, 0, stream>>>(points, xyz, pts_t);

  const size_t shA = (size_t)(3 * N_ + 64 * CIN + 3 * 64) * sizeof(float) +
                     (size_t)256 * KNBR * sizeof(int);              // ~100 KB
  ball_mlp_k<<<B_ * (N_ / 256), 256, shA, stream>>>(xyz, pts_t, W, bb, g, bt,
                                                    rm, rv, feat);

  const size_t shB = (size_t)(P1_ * 66 + P2_ * 130 + 2 * P1_ + 2 * P2_ +
                              8 * 16 * 132) * sizeof(float);        // ~133 KB
  resnet_k<<<B_ * (N_ / 128), 256, shB, stream>>>(feat, points, W1, b1, g1, bt1,
                                                  rm1, rv1, W2, b2, g2, bt2,
                                                  rm2, rv2,
                                                  out + (size_t)B_ * 3 * N_);
}